// encoderNet_21758304322145
// MI455X (gfx1250) — compile-verified
//
#include <hip/hip_runtime.h>
#include <hip/hip_bf16.h>

// ---------------- problem constants (from reference) ----------------
#define NEG_SLOPE 0.01f
#define BATCH   64
#define NNODES  16384              // N = 128*128
#define BN      (BATCH * NNODES)   // 1,048,576
#define DEG_E   8
#define NEDGES  (BN * DEG_E)       // 8,388,608
#define C_OUT   8
#define K1      (NNODES * C_OUT)   // 131072  (fc1 reduction dim)
#define DIM     128
#define B_OUT   16
#define KCHUNK  1024               // split-K chunk for fc1 (multiple of 8)
#define KCHUNKS (K1 / KCHUNK)      // 128

typedef __attribute__((ext_vector_type(2))) float v2f;
typedef __attribute__((ext_vector_type(8))) float v8f;

__device__ __forceinline__ float leaky(float v) {
    // leaky_relu(v, 0.01) == max(v, 0.01*v) for slope < 1
    return fmaxf(v, NEG_SLOPE * v);
}

// ---------------- phase 0: zero workspace ----------------
__global__ void k_zero(float* __restrict__ p, int n) {
    int i = blockIdx.x * blockDim.x + threadIdx.x;
    if (i < n) p[i] = 0.0f;
}

// ---------------- phase 1: weighted degree ----------------
// edge streams are read-once -> non-temporal (protect L2 for deg/x/tx1)
__global__ void k_deg(const int* __restrict__ row,
                      const float* __restrict__ ea,
                      float* __restrict__ deg) {
    int e = blockIdx.x * blockDim.x + threadIdx.x;
    if (e < NEDGES) {
        int   r = __builtin_nontemporal_load(&row[e]);
        float a = __builtin_nontemporal_load(&ea[e]);
        atomicAdd(&deg[r], a);
    }
}

// ---------------- phase 2: normalized edge scatter (Tx1) ----------------
__global__ void k_edge(const int* __restrict__ row,
                       const int* __restrict__ col,
                       const float* __restrict__ ea,
                       const float* __restrict__ x,
                       const float* __restrict__ deg,
                       float* __restrict__ tx1) {
    int e = blockIdx.x * blockDim.x + threadIdx.x;
    if (e >= NEDGES) return;
    int   r = __builtin_nontemporal_load(&row[e]);
    int   c = __builtin_nontemporal_load(&col[e]);
    float a = __builtin_nontemporal_load(&ea[e]);
    float dr = deg[r];                       // L2-resident gathers stay RT
    float dc = deg[c];
    float ir = (dr > 0.0f) ? __frsqrt_rn(dr) : 0.0f;
    float ic = (dc > 0.0f) ? __frsqrt_rn(dc) : 0.0f;
    float w = -ir * a * ic;
    atomicAdd(&tx1[r], w * x[c]);
}

// ---------------- phase 3: fc1 split-K GEMM via V_WMMA_F32_16X16X4_F32 -----
// grid = (8 n-tiles, KCHUNKS); block = 32 (one wave). M=64 -> 4 m-tiles/wave.
// K-loop steps by 8: one graph node per group; x/tx1 gathered ONCE per
// (m, node) and expanded to 8 channels in registers (2 WMMA k-steps).
__global__ void k_fc1(const float* __restrict__ x,
                      const float* __restrict__ tx1,
                      const float* __restrict__ W0,
                      const float* __restrict__ W1,
                      const float* __restrict__ cb,
                      const float* __restrict__ fc1W,   // [K1, DIM] row-major
                      float* __restrict__ acc1) {       // [64, DIM] split-K accumulator
    const int lane  = threadIdx.x;          // 0..31, EXEC all ones
    const int n0    = blockIdx.x * 16;      // column tile base
    const int k0    = blockIdx.y * KCHUNK;  // K chunk base
    const int mrow  = lane & 15;            // M (A) / N (B,C) sub-index
    const int khalf = (lane >> 4) * 2;      // upper half lanes hold K+2,K+3
    const int bcol  = n0 + mrow;

    // channels this lane touches within each 8-wide node group (loop-invariant)
    const int ch0 = khalf, ch1 = khalf + 1, ch2 = khalf + 4, ch3 = khalf + 5;
    const float w00 = W0[ch0], w01 = W0[ch1], w02 = W0[ch2], w03 = W0[ch3];
    const float w10 = W1[ch0], w11 = W1[ch1], w12 = W1[ch2], w13 = W1[ch3];
    const float bb0 = cb[ch0], bb1 = cb[ch1], bb2 = cb[ch2], bb3 = cb[ch3];

    v8f d[4] = {v8f{}, v8f{}, v8f{}, v8f{}};

    for (int k = k0; k < k0 + KCHUNK; k += 8) {
        const int node = k >> 3;
        // B fragments for the two WMMA k-steps (rows k+khalf.., streamed NT)
        const float* bp = fc1W + (size_t)(k + khalf) * DIM + bcol;
        v2f bfA, bfB;
        bfA.x = __builtin_nontemporal_load(bp);
        bfA.y = __builtin_nontemporal_load(bp + DIM);
        bfB.x = __builtin_nontemporal_load(bp + 4 * DIM);
        bfB.y = __builtin_nontemporal_load(bp + 5 * DIM);
#pragma unroll
        for (int mt = 0; mt < 4; ++mt) {
            const int g = (mt * 16 + mrow) * NNODES + node;  // L2-resident gather
            const float xv = x[g];
            const float tv = tx1[g];
            v2f aA, aB;
            aA.x = leaky(fmaf(xv, w00, fmaf(tv, w10, bb0)));
            aA.y = leaky(fmaf(xv, w01, fmaf(tv, w11, bb1)));
            aB.x = leaky(fmaf(xv, w02, fmaf(tv, w12, bb2)));
            aB.y = leaky(fmaf(xv, w03, fmaf(tv, w13, bb3)));
            d[mt] = __builtin_amdgcn_wmma_f32_16x16x4_f32(false, aA, false, bfA, (short)0, d[mt], false, false);
            d[mt] = __builtin_amdgcn_wmma_f32_16x16x4_f32(false, aB, false, bfB, (short)0, d[mt], false, false);
        }
    }

    // C/D layout: vgpr r, lanes 0-15 -> M=r, lanes 16-31 -> M=r+8; N = n0 + (lane&15)
    const int cn   = n0 + (lane & 15);
    const int mofs = (lane >> 4) * 8;
#pragma unroll
    for (int mt = 0; mt < 4; ++mt) {
#pragma unroll
        for (int r = 0; r < 8; ++r) {
            atomicAdd(&acc1[(mt * 16 + mofs + r) * DIM + cn], d[mt][r]);
        }
    }
}

// ---------------- phase 4: fc2 (64x128 @ 128x128), bias+leaky on both sides -
// one block of 256 threads = 8 waves, wave w handles n-tile w
__global__ void k_fc2(const float* __restrict__ acc1,
                      const float* __restrict__ fc1b,
                      const float* __restrict__ fc2W,  // [DIM, DIM]
                      const float* __restrict__ fc2b,
                      float* __restrict__ h2) {        // [64, DIM]
    const int lane  = threadIdx.x & 31;
    const int n0    = (threadIdx.x >> 5) * 16;
    const int mrow  = lane & 15;
    const int khalf = (lane >> 4) * 2;

    v8f d[4] = {v8f{}, v8f{}, v8f{}, v8f{}};

    for (int k = 0; k < DIM; k += 4) {
        const int ka = k + khalf;
        const int kb = ka + 1;
        v2f bf;
        bf.x = fc2W[ka * DIM + n0 + mrow];
        bf.y = fc2W[kb * DIM + n0 + mrow];
        const float b1a = fc1b[ka];
        const float b1b = fc1b[kb];
#pragma unroll
        for (int mt = 0; mt < 4; ++mt) {
            v2f a;
            float t0 = acc1[(mt * 16 + mrow) * DIM + ka] + b1a;
            float t1 = acc1[(mt * 16 + mrow) * DIM + kb] + b1b;
            a.x = leaky(t0);
            a.y = leaky(t1);
            d[mt] = __builtin_amdgcn_wmma_f32_16x16x4_f32(false, a, false, bf, (short)0, d[mt], false, false);
        }
    }

    const int cn   = n0 + (lane & 15);
    const int mofs = (lane >> 4) * 8;
    const float bb = fc2b[cn];
#pragma unroll
    for (int mt = 0; mt < 4; ++mt) {
#pragma unroll
        for (int r = 0; r < 8; ++r) {
            h2[(mt * 16 + mofs + r) * DIM + cn] = leaky(d[mt][r] + bb);
        }
    }
}

// ---------------- phase 5: fc3 (64x128 @ 128x16) + bias, no activation ------
// single wave
__global__ void k_fc3(const float* __restrict__ h2,
                      const float* __restrict__ fc3W,  // [DIM, B_OUT]
                      const float* __restrict__ fc3b,
                      float* __restrict__ out) {       // [64, B_OUT]
    const int lane  = threadIdx.x;
    const int mrow  = lane & 15;
    const int khalf = (lane >> 4) * 2;

    v8f d[4] = {v8f{}, v8f{}, v8f{}, v8f{}};

    for (int k = 0; k < DIM; k += 4) {
        const int ka = k + khalf;
        const int kb = ka + 1;
        v2f bf;
        bf.x = fc3W[ka * B_OUT + mrow];
        bf.y = fc3W[kb * B_OUT + mrow];
#pragma unroll
        for (int mt = 0; mt < 4; ++mt) {
            v2f a;
            a.x = h2[(mt * 16 + mrow) * DIM + ka];
            a.y = h2[(mt * 16 + mrow) * DIM + kb];
            d[mt] = __builtin_amdgcn_wmma_f32_16x16x4_f32(false, a, false, bf, (short)0, d[mt], false, false);
        }
    }

    const int cn   = lane & 15;
    const int mofs = (lane >> 4) * 8;
    const float bb = fc3b[cn];
#pragma unroll
    for (int mt = 0; mt < 4; ++mt) {
#pragma unroll
        for (int r = 0; r < 8; ++r) {
            out[(mt * 16 + mofs + r) * B_OUT + cn] = d[mt][r] + bb;
        }
    }
}

// ---------------- host launcher ----------------
extern "C" void kernel_launch(void* const* d_in, const int* in_sizes, int n_in,
                              void* d_out, int out_size, void* d_ws, size_t ws_size,
                              hipStream_t stream) {
    (void)in_sizes; (void)n_in; (void)out_size; (void)ws_size;

    const float* x    = (const float*)d_in[0];           // [B, N] flat = BN
    const int*   eidx = (const int*)  d_in[1];           // [2, E]
    const float* ea   = (const float*)d_in[2];           // [E]
    const float* W0   = (const float*)d_in[3];           // [1, 8]
    const float* W1   = (const float*)d_in[4];           // [1, 8]
    const float* cb   = (const float*)d_in[5];           // [8]
    const float* f1W  = (const float*)d_in[6];           // [131072, 128]
    const float* f1b  = (const float*)d_in[7];           // [128]
    const float* f2W  = (const float*)d_in[8];           // [128, 128]
    const float* f2b  = (const float*)d_in[9];           // [128]
    const float* f3W  = (const float*)d_in[10];          // [128, 16]
    const float* f3b  = (const float*)d_in[11];          // [16]
    float*       out  = (float*)d_out;                   // [64, 1, 16]

    const int* row = eidx;
    const int* col = eidx + NEDGES;

    float* ws   = (float*)d_ws;
    float* deg  = ws;                       // BN floats
    float* tx1  = ws + BN;                  // BN floats
    float* acc1 = ws + 2 * (size_t)BN;      // 64*128 floats
    float* h2   = acc1 + BATCH * DIM;       // 64*128 floats

    // zero deg + tx1 + acc1
    const int nz = 2 * BN + BATCH * DIM;
    k_zero<<<(nz + 255) / 256, 256, 0, stream>>>(ws, nz);

    // weighted degree
    k_deg<<<(NEDGES + 255) / 256, 256, 0, stream>>>(row, ea, deg);

    // normalized edge scatter -> Tx1
    k_edge<<<(NEDGES + 255) / 256, 256, 0, stream>>>(row, col, ea, x, deg, tx1);

    // fc1 split-K WMMA GEMM (ChebConv feature fused into A operand)
    k_fc1<<<dim3(DIM / 16, KCHUNKS), 32, 0, stream>>>(x, tx1, W0, W1, cb, f1W, acc1);

    // fc2 (one workgroup, 8 waves)
    k_fc2<<<1, 256, 0, stream>>>(acc1, f1b, f2W, f2b, h2);

    // fc3 (one wave)
    k_fc3<<<1, 32, 0, stream>>>(h2, f3W, f3b, out);
}